// Single_Net_14147622273473
// MI455X (gfx1250) — compile-verified
//
#include <hip/hip_runtime.h>

typedef __attribute__((ext_vector_type(2))) float v2f;
typedef __attribute__((ext_vector_type(8))) float v8f;

#define F_IN  256
#define F_OUT 64

// ---------------- degree kernels ----------------

__global__ __launch_bounds__(256) void deg_init_kernel(float* __restrict__ deg, int n) {
    int i = blockIdx.x * 256 + threadIdx.x;
    if (i < n) deg[i] = 1.0f;   // self-loop weight
}

__global__ __launch_bounds__(256) void deg_accum_kernel(float* __restrict__ deg,
                                                        const int* __restrict__ dst,
                                                        const float* __restrict__ ew,
                                                        int e) {
    int i = blockIdx.x * 256 + threadIdx.x;
    if (i < e) unsafeAtomicAdd(&deg[dst[i]], ew[i]);
}

__global__ __launch_bounds__(256) void deg_rsqrt_kernel(float* __restrict__ deg, int n) {
    int i = blockIdx.x * 256 + threadIdx.x;
    if (i < n) {
        float v = deg[i];
        deg[i] = (v > 0.0f) ? rsqrtf(v) : 0.0f;
    }
}

// ---------------- GEMM: h = x @ W  via fp32 WMMA 16x16x4 ----------------
// One wave computes a 16x64 output slab (4 N-tiles of 16x16).
// A lane layout: a[v] = x[rowBase + (lane&15)][4*kk + 2*(lane>>4) + v]
// B is staged in LDS pre-swizzled as sB[kk][khalf][n][v] (v = K parity pair)
// so each lane's B operand for tile nt is one contiguous ds_load_b64 at
//   sB + kk*256 + khalf*128 + (nt*16 + m)*2.
// C/D: VGPR v -> row (v + 8*(lane>>4)), col (lane&15) within tile.

__global__ __launch_bounds__(256) void gemm_wmma_kernel(const float* __restrict__ x,
                                                        const float* __restrict__ w,
                                                        float* __restrict__ h,
                                                        int nNodes) {
    __shared__ float sB[F_IN * F_OUT];   // 64 KB of the 320 KB WGP LDS, swizzled

    // Coalesced 128-bit reads of W, swizzled scatter into LDS.
    // Linear element l = k*64 + n  ->  sB[(k>>2)*256 + ((k>>1)&1)*128 + n*2 + (k&1)]
    {
        const float4* w4 = (const float4*)w;
        for (int i4 = threadIdx.x; i4 < (F_IN * F_OUT) / 4; i4 += 256) {
            float4 val = w4[i4];
            int l = i4 * 4;
            int k = l >> 6;
            int n = l & 63;                       // n, n+1, n+2, n+3 share k
            float* dstp = &sB[(k >> 2) * 256 + ((k >> 1) & 1) * 128 + (k & 1) + n * 2];
            dstp[0] = val.x;
            dstp[2] = val.y;
            dstp[4] = val.z;
            dstp[6] = val.w;
        }
    }
    __syncthreads();

    const int wave  = threadIdx.x >> 5;
    const int lane  = threadIdx.x & 31;
    const int m     = lane & 15;
    const int khalf = lane >> 4;

    const int rowTile = blockIdx.x * 8 + wave;
    const int rowBase = rowTile * 16;
    if (rowBase >= nNodes) return;          // wave-uniform: EXEC stays all-1s

    int row = rowBase + m;
    if (row >= nNodes) row = nNodes - 1;    // clamp loads, keep EXEC full

    const float* __restrict__ xrow = x + (size_t)row * F_IN;
    const float* __restrict__ bbase = &sB[khalf * 128 + m * 2];

    v8f acc0 = {}, acc1 = {}, acc2 = {}, acc3 = {};

    #pragma unroll 4
    for (int kk = 0; kk < F_IN / 4; ++kk) {
        const int k0 = kk * 4 + khalf * 2;

        // x streamed exactly once: non-temporal load (TH=NT), don't pollute L2
        v2f a = __builtin_nontemporal_load((const v2f*)(xrow + k0));

        const float* bp = bbase + kk * 256;          // 8B-aligned contiguous pairs
        v2f b0 = *(const v2f*)(bp +  0);
        v2f b1 = *(const v2f*)(bp + 32);
        v2f b2 = *(const v2f*)(bp + 64);
        v2f b3 = *(const v2f*)(bp + 96);

        acc0 = __builtin_amdgcn_wmma_f32_16x16x4_f32(false, a, false, b0, (short)0, acc0, false, false);
        acc1 = __builtin_amdgcn_wmma_f32_16x16x4_f32(false, a, false, b1, (short)0, acc1, false, false);
        acc2 = __builtin_amdgcn_wmma_f32_16x16x4_f32(false, a, false, b2, (short)0, acc2, false, false);
        acc3 = __builtin_amdgcn_wmma_f32_16x16x4_f32(false, a, false, b3, (short)0, acc3, false, false);
    }

    float* hp0 = h + (size_t)(rowBase + khalf * 8) * F_OUT + m;
    if (rowBase + 16 <= nNodes) {
        // full tile in range: straight-line stores, no exec juggling
        #pragma unroll
        for (int v = 0; v < 8; ++v) {
            float* hp = hp0 + (size_t)v * F_OUT;
            hp[0]  = acc0[v];
            hp[16] = acc1[v];
            hp[32] = acc2[v];
            hp[48] = acc3[v];
        }
    } else {
        #pragma unroll
        for (int v = 0; v < 8; ++v) {
            if (rowBase + khalf * 8 + v < nNodes) {
                float* hp = hp0 + (size_t)v * F_OUT;
                hp[0]  = acc0[v];
                hp[16] = acc1[v];
                hp[32] = acc2[v];
                hp[48] = acc3[v];
            }
        }
    }
}

// ---------------- self-loop + bias init of out ----------------
// out[n][f] = bias[f] + dis[n]^2 * h[n][f]

__global__ __launch_bounds__(256) void selfloop_bias_kernel(const float* __restrict__ h,
                                                            const float* __restrict__ dis,
                                                            const float* __restrict__ bias,
                                                            float* __restrict__ out,
                                                            int nNodes) {
    int idx = blockIdx.x * 256 + threadIdx.x;
    if (idx >= nNodes * F_OUT) return;
    int n = idx >> 6;
    int f = idx & 63;
    float d = dis[n];
    out[idx] = bias[f] + d * d * h[idx];
}

// ---------------- edge scatter: one edge per wave32 ----------------
// lane handles features (2*lane, 2*lane+1): float2 gather of h[src],
// two native global_atomic_add_f32 into out[dst].

__global__ __launch_bounds__(256) void scatter_kernel(const float* __restrict__ h,
                                                      const float* __restrict__ dis,
                                                      const int* __restrict__ srcIdx,
                                                      const int* __restrict__ dstIdx,
                                                      const float* __restrict__ ew,
                                                      float* __restrict__ out,
                                                      int nEdges) {
    int e    = (blockIdx.x * 256 + threadIdx.x) >> 5;
    int lane = threadIdx.x & 31;
    if (e >= nEdges) return;

    int s = srcIdx[e];
    int d = dstIdx[e];
    float norm = dis[s] * ew[e] * dis[d];

    v2f hv = *(const v2f*)(h + (size_t)s * F_OUT + 2 * lane);
    float* op = out + (size_t)d * F_OUT + 2 * lane;
    unsafeAtomicAdd(op + 0, norm * hv[0]);
    unsafeAtomicAdd(op + 1, norm * hv[1]);
}

// ---------------- launcher ----------------

extern "C" void kernel_launch(void* const* d_in, const int* in_sizes, int n_in,
                              void* d_out, int out_size, void* d_ws, size_t ws_size,
                              hipStream_t stream) {
    const float* x    = (const float*)d_in[0];   // [N, 256]
    const float* w    = (const float*)d_in[1];   // [256, 64]
    const float* bias = (const float*)d_in[2];   // [64]
    const int*   ei   = (const int*)d_in[3];     // [2, E] (JAX default x64-off -> int32)
    const float* ew   = (const float*)d_in[4];   // [E]

    const int N = in_sizes[0] / F_IN;
    const int E = in_sizes[4];

    float* out = (float*)d_out;

    // workspace layout: [deg/dis : N floats][pad][h : N*64 floats]
    float* deg = (float*)d_ws;
    size_t hOff = ((size_t)N + 255) & ~(size_t)255;
    float* h = (float*)d_ws + hOff;

    const int* srcIdx = ei;
    const int* dstIdx = ei + E;

    // 1) deg = 1 (self loop), += w per in-edge, then deg <- deg^-1/2
    deg_init_kernel<<<(N + 255) / 256, 256, 0, stream>>>(deg, N);
    deg_accum_kernel<<<(E + 255) / 256, 256, 0, stream>>>(deg, dstIdx, ew, E);
    deg_rsqrt_kernel<<<(N + 255) / 256, 256, 0, stream>>>(deg, N);

    // 2) h = x @ W  (fp32 WMMA), 8 row-tiles of 16 per block
    int rowTiles = (N + 15) / 16;
    int gemmBlocks = (rowTiles + 7) / 8;
    gemm_wmma_kernel<<<gemmBlocks, 256, 0, stream>>>(x, w, h, N);

    // 3) out = bias + dis^2 * h   (self-loop term, also initializes out)
    selfloop_bias_kernel<<<((size_t)N * F_OUT + 255) / 256, 256, 0, stream>>>(h, deg, bias, out, N);

    // 4) out[dst] += dis[src]*w*dis[dst] * h[src]  (one edge per wave)
    int scatterBlocks = (E + 7) / 8;   // 8 waves (edges) per 256-thread block
    scatter_kernel<<<scatterBlocks, 256, 0, stream>>>(h, deg, srcIdx, dstIdx, ew, out, E);
}